// Model_54760833024242
// MI455X (gfx1250) — compile-verified
//
#include <hip/hip_runtime.h>
#include <cstddef>

typedef float v2f __attribute__((ext_vector_type(2)));
typedef float v8f __attribute__((ext_vector_type(8)));

#define NN 64
#define CC 64
#define TT 300
#define VV 25
#define DD 64
#define NT (NN*TT)            /* 19200 */
#define X0_SZ (NN*CC*TT*VV)   /* 30,720,000 */
#define L_SUM (TT+VV)         /* 325 */
#define D_BLOCKS 300
#define D_WAVES (D_BLOCKS*4)  /* 1200 waves; 19200/1200 = 16 tiles per wave */

__device__ __forceinline__ float hswish_f(float y) {
    return y * fminf(fmaxf(y + 3.0f, 0.0f), 6.0f) * (1.0f/6.0f);
}
__device__ __forceinline__ float sigmoid_f(float x) {
    return 1.0f / (1.0f + __expf(-x));
}
__device__ __forceinline__ v8f v8_splat(float x) {
    v8f r;
    #pragma unroll
    for (int j = 0; j < 8; ++j) r[j] = x;
    return r;
}

// ---------------------------------------------------------------------------
// Phase A: per-(n,c) slice: tconv[v] = sum_t Wt[t]*x0[n,c,t,(v+c)%V] + bt
//                           vconv[t] = sum_v Wv[v]*x0[n,c,t,(v+c)%V] + bv
// plus block partial sums for the two global (single-channel) batchnorms.
// ---------------------------------------------------------------------------
__global__ __launch_bounds__(256) void kA(
    const float* __restrict__ x0, const float* __restrict__ Wt, const float* __restrict__ bt,
    const float* __restrict__ Wv, const float* __restrict__ bv,
    float* __restrict__ tconv, float* __restrict__ vconv, float* __restrict__ partA)
{
    __shared__ float sx[TT*VV];
    __shared__ float swt[TT];
    __shared__ float swv[VV];
    __shared__ float part[8][32];
    __shared__ float red[256];
    int blk = blockIdx.x;          // blk = n*64 + c
    int c = blk & 63;
    int tid = threadIdx.x;
    const float* xs = x0 + (size_t)blk * (TT*VV);
    for (int i = tid; i < TT*VV; i += 256) sx[i] = xs[i];
    for (int i = tid; i < TT; i += 256) swt[i] = Wt[i];
    if (tid < VV) swv[tid] = Wv[(tid - (c % VV) + VV) % VV];  // rotated Wv
    __syncthreads();

    float bvv = bv[0], btt = bt[0];
    float sum_v = 0.f, sq_v = 0.f;
    for (int t = tid; t < TT; t += 256) {
        float acc = bvv;
        #pragma unroll
        for (int w = 0; w < VV; ++w) acc += swv[w] * sx[t*VV + w];
        vconv[(size_t)blk*TT + t] = acc;
        sum_v += acc; sq_v += acc*acc;
    }
    int w = tid & 31, g = tid >> 5;
    float a = 0.f;
    if (w < VV) { for (int t = g; t < TT; t += 8) a += swt[t] * sx[t*VV + w]; }
    part[g][w] = a;
    __syncthreads();
    float sum_t = 0.f, sq_t = 0.f;
    if (tid < VV) {
        float s = 0.f;
        #pragma unroll
        for (int gg = 0; gg < 8; ++gg) s += part[gg][tid];
        part[0][tid] = s;          // tmp[w] = sum_t Wt[t]*x0[..,t,w]
    }
    __syncthreads();
    if (tid < VV) {
        int v = tid;
        float val = part[0][(v + c) % VV] + btt;
        tconv[(size_t)blk*VV + v] = val;
        sum_t = val; sq_t = val*val;
    }
    float vals[4] = {sum_t, sq_t, sum_v, sq_v};
    #pragma unroll
    for (int q = 0; q < 4; ++q) {
        __syncthreads();
        red[tid] = vals[q];
        __syncthreads();
        for (int off = 128; off > 0; off >>= 1) {
            if (tid < off) red[tid] += red[tid + off];
            __syncthreads();
        }
        if (tid == 0) partA[(size_t)blk*4 + q] = red[0];
    }
}

__global__ __launch_bounds__(256) void kA2(const float* __restrict__ partA,
                                           float* __restrict__ statsA)
{
    __shared__ float red[256][4];
    int tid = threadIdx.x;
    float s[4] = {0.f, 0.f, 0.f, 0.f};
    for (int b = tid; b < 4096; b += 256)
        for (int q = 0; q < 4; ++q) s[q] += partA[(size_t)b*4 + q];
    for (int q = 0; q < 4; ++q) red[tid][q] = s[q];
    __syncthreads();
    for (int off = 128; off > 0; off >>= 1) {
        if (tid < off)
            for (int q = 0; q < 4; ++q) red[tid][q] += red[tid+off][q];
        __syncthreads();
    }
    if (tid == 0) {
        float cnt_t = (float)(NN*CC*VV);
        float cnt_v = (float)(NN*CC*TT);
        float mt = red[0][0] / cnt_t;
        float vt = red[0][1] / cnt_t - mt*mt;
        float mv = red[0][2] / cnt_v;
        float vv2 = red[0][3] / cnt_v - mv*mv;
        statsA[0] = mt; statsA[1] = rsqrtf(vt + 1e-5f);
        statsA[2] = mv; statsA[3] = rsqrtf(vv2 + 1e-5f);
    }
}

// ---------------------------------------------------------------------------
// Phase B: xs_raw[n,d,l] = sum_c Ws[d,c]*S[n,c,l] + bs[d]    (S = bn+hswish of
// the concatenated/reshaped tconv|vconv plane), then per-d stats.
// ---------------------------------------------------------------------------
__global__ __launch_bounds__(256) void kB(
    const float* __restrict__ tconv, const float* __restrict__ vconv,
    const float* __restrict__ statsA,
    const float* __restrict__ g_t, const float* __restrict__ be_t,
    const float* __restrict__ g_v, const float* __restrict__ be_v,
    const float* __restrict__ Ws, const float* __restrict__ bs,
    float* __restrict__ xs_raw)
{
    __shared__ float S[64][16];
    int l0 = blockIdx.x * 16;
    int n = blockIdx.y;
    int tid = threadIdx.x;
    float mt = statsA[0], it = statsA[1], mv = statsA[2], iv = statsA[3];
    float gt = g_t[0], bet = be_t[0], gv = g_v[0], bev = be_v[0];
    for (int idx = tid; idx < 64*16; idx += 256) {
        int cch = idx >> 4, li = idx & 15, l = l0 + li;
        float val = 0.f;
        if (l < L_SUM) {
            int k = cch * L_SUM + l;
            float y;
            if (k < CC*VV) {
                int ct = k / VV, v = k % VV;
                y = tconv[(size_t)(n*CC + ct)*VV + v];
                y = (y - mt) * it * gt + bet;
            } else {
                int k2 = k - CC*VV;
                int cv = k2 / TT, t = k2 % TT;
                y = vconv[(size_t)(n*CC + cv)*TT + t];
                y = (y - mv) * iv * gv + bev;
            }
            val = hswish_f(y);
        }
        S[cch][li] = val;
    }
    __syncthreads();
    int d = tid >> 2;
    int li0 = (tid & 3) * 4;
    float b0 = bs[d];
    float acc0 = b0, acc1 = b0, acc2 = b0, acc3 = b0;
    for (int cch = 0; cch < 64; ++cch) {
        float wv = Ws[d*64 + cch];
        acc0 += wv * S[cch][li0+0];
        acc1 += wv * S[cch][li0+1];
        acc2 += wv * S[cch][li0+2];
        acc3 += wv * S[cch][li0+3];
    }
    float accs[4] = {acc0, acc1, acc2, acc3};
    #pragma unroll
    for (int q = 0; q < 4; ++q) {
        int l = l0 + li0 + q;
        if (l < L_SUM) xs_raw[(size_t)(n*64 + d)*L_SUM + l] = accs[q];
    }
}

__global__ __launch_bounds__(256) void kB2(const float* __restrict__ xs_raw,
                                           float* __restrict__ statsB)
{
    __shared__ float redS[256], redQ[256];
    int d = blockIdx.x, tid = threadIdx.x;
    float s = 0.f, q = 0.f;
    for (int e = tid; e < NN * L_SUM; e += 256) {
        int n = e / L_SUM, l = e % L_SUM;
        float y = xs_raw[(size_t)(n*64 + d)*L_SUM + l];
        s += y; q += y*y;
    }
    redS[tid] = s; redQ[tid] = q;
    __syncthreads();
    for (int off = 128; off > 0; off >>= 1) {
        if (tid < off) { redS[tid] += redS[tid+off]; redQ[tid] += redQ[tid+off]; }
        __syncthreads();
    }
    if (tid == 0) {
        float cnt = (float)(NN * L_SUM);
        float m = redS[0] / cnt;
        float var = redQ[0] / cnt - m*m;
        statsB[d*2]   = m;
        statsB[d*2+1] = rsqrtf(var + 1e-5f);
    }
}

// ---------------------------------------------------------------------------
// Phase C: m[d,l] = mean_n hswish(bn(xs_raw)); then the two folded matmuls.
// ---------------------------------------------------------------------------
__global__ __launch_bounds__(64) void kC1(
    const float* __restrict__ xs_raw, const float* __restrict__ statsB,
    const float* __restrict__ g_s, const float* __restrict__ be_s,
    float* __restrict__ mbuf)
{
    int l = blockIdx.x, d = threadIdx.x;
    float m = statsB[d*2], inv = statsB[d*2+1], g = g_s[d], be = be_s[d];
    float s = 0.f;
    for (int n = 0; n < NN; ++n) {
        float y = xs_raw[(size_t)(n*64 + d)*L_SUM + l];
        s += hswish_f((y - m) * inv * g + be);
    }
    mbuf[d*L_SUM + l] = s * (1.0f/NN);
}

__global__ __launch_bounds__(256) void kC2(
    const float* __restrict__ mbuf, const float* __restrict__ W2t,
    const float* __restrict__ b2t, float* __restrict__ out_time)
{
    int idx = blockIdx.x * 256 + threadIdx.x;      // exactly 19200
    int d = idx / TT, t = idx % TT;
    float acc = b2t[d];
    for (int dd = 0; dd < 64; ++dd) acc += W2t[d*64+dd] * mbuf[dd*L_SUM + t];
    out_time[d*TT + t] = sigmoid_f(acc);
}

__global__ __launch_bounds__(256) void kC3(
    const float* __restrict__ mbuf, const float* __restrict__ W2v,
    const float* __restrict__ b2v, float* __restrict__ scaleb)
{
    for (int i = threadIdx.x; i < VV*64; i += 256) {
        int v = i / 64, cch = i % 64;
        float acc = b2v[cch];
        for (int dd = 0; dd < 64; ++dd) acc += W2v[cch*64+dd] * mbuf[dd*L_SUM + TT + v];
        scaleb[v*64 + cch] = tanhf(sigmoid_f(acc)) + 1.0f;
    }
}

// ---------------------------------------------------------------------------
// Phase D core: per wave, one (n,t) tile. A(25x64, padded to 32) built from the
// LDS x0 slice with fused (v+c)%V shift and attention scale; B = Lw (LDS).
// Z = A@Lw + Lb via V_WMMA_F32_16X16X4_F32 chains (2 Mtiles x 4 Ntiles x 16 K).
// ---------------------------------------------------------------------------
__device__ __forceinline__ void compute_tile(
    const float* xw, const float* sLw, const float* sScale,
    int laneM, int half, const float lb[4], v8f acc[2][4])
{
    #pragma unroll
    for (int i = 0; i < 2; ++i)
        #pragma unroll
        for (int jn = 0; jn < 4; ++jn) acc[i][jn] = v8_splat(lb[jn]);

    #pragma unroll
    for (int kk = 0; kk < 16; ++kk) {
        int c0 = 4*kk + 2*half;                  // A/B K-slots per ISA layout
        v2f af[2];
        #pragma unroll
        for (int i = 0; i < 2; ++i) {
            int v = 16*i + laneM;                // A row (lanes 0-15 / 16-31 same M)
            if (v < VV) {
                af[i].x = xw[c0*VV     + (v + c0    ) % VV] * sScale[v*64 + c0    ];
                af[i].y = xw[(c0+1)*VV + (v + c0 + 1) % VV] * sScale[v*64 + c0 + 1];
            } else {
                af[i].x = 0.f; af[i].y = 0.f;    // padded rows
            }
        }
        v2f bf[4];
        #pragma unroll
        for (int jn = 0; jn < 4; ++jn) {
            int col = 16*jn + laneM;
            bf[jn].x = sLw[c0*64       + col];
            bf[jn].y = sLw[(c0+1)*64   + col];
        }
        #pragma unroll
        for (int i = 0; i < 2; ++i)
            #pragma unroll
            for (int jn = 0; jn < 4; ++jn)
                acc[i][jn] = __builtin_amdgcn_wmma_f32_16x16x4_f32(
                    false, af[i], false, bf[jn], (short)0, acc[i][jn], false, false);
    }
}

// D1: column sums / sum-of-squares of Z accumulated in the C-register slots
// (each slot is a fixed (v,d) column across all nt iterations of a wave).
__global__ __launch_bounds__(128) void kD1(
    const float* __restrict__ x0, const float* __restrict__ Lw, const float* __restrict__ Lb,
    const float* __restrict__ scaleb, float* __restrict__ partZ)
{
    __shared__ float sLw[64*64];
    __shared__ float sScale[VV*64];
    __shared__ float sX[4][VV*64];
    int tid = threadIdx.x;
    for (int i = tid; i < 64*64; i += 128) sLw[i] = Lw[i];
    for (int i = tid; i < VV*64; i += 128) sScale[i] = scaleb[i];
    __syncthreads();
    int wave = tid >> 5, lane = tid & 31;
    int laneM = lane & 15, half = lane >> 4;
    float* xw = sX[wave];
    int waveg = blockIdx.x * 4 + wave;
    float lb[4];
    #pragma unroll
    for (int jn = 0; jn < 4; ++jn) lb[jn] = Lb[16*jn + laneM];

    v8f Sacc[2][4], Qacc[2][4];
    #pragma unroll
    for (int i = 0; i < 2; ++i)
        #pragma unroll
        for (int jn = 0; jn < 4; ++jn) { Sacc[i][jn] = v8_splat(0.f); Qacc[i][jn] = v8_splat(0.f); }

    for (int it = 0; it < 16; ++it) {
        int nt = waveg + it * D_WAVES;
        int n = nt / TT, t = nt % TT;
        const float* xsrc = x0 + (size_t)n * (CC*TT*VV) + (size_t)t * VV;
        if (it + 1 < 16) {
            int nt2 = waveg + (it+1) * D_WAVES;
            __builtin_prefetch(x0 + (size_t)(nt2/TT)*(CC*TT*VV) + (size_t)(nt2%TT)*VV, 0, 1);
        }
        for (int i2 = lane; i2 < CC*VV; i2 += 32) {
            int cch = i2 / VV, v = i2 % VV;
            xw[i2] = xsrc[(size_t)cch * (TT*VV) + v];
        }
        __syncthreads();
        v8f acc[2][4];
        compute_tile(xw, sLw, sScale, laneM, half, lb, acc);
        __syncthreads();
        #pragma unroll
        for (int i = 0; i < 2; ++i)
            #pragma unroll
            for (int jn = 0; jn < 4; ++jn) {
                Sacc[i][jn] += acc[i][jn];
                Qacc[i][jn] += acc[i][jn] * acc[i][jn];
            }
    }
    #pragma unroll
    for (int i = 0; i < 2; ++i)
        #pragma unroll
        for (int jn = 0; jn < 4; ++jn)
            #pragma unroll
            for (int j = 0; j < 8; ++j) {
                int v = 16*i + j + 8*half;           // C layout: VGPR j -> M=j / j+8
                if (v < VV) {
                    int col = v*64 + 16*jn + laneM;
                    partZ[((size_t)waveg*1600 + col)*2 + 0] = Sacc[i][jn][j];
                    partZ[((size_t)waveg*1600 + col)*2 + 1] = Qacc[i][jn][j];
                }
            }
}

__global__ __launch_bounds__(256) void kD2(const float* __restrict__ partZ,
                                           float* __restrict__ zstats)
{
    int col = blockIdx.x * 256 + threadIdx.x;
    if (col >= 1600) return;
    float s = 0.f, q = 0.f;
    for (int w = 0; w < D_WAVES; ++w) {
        s += partZ[((size_t)w*1600 + col)*2 + 0];
        q += partZ[((size_t)w*1600 + col)*2 + 1];
    }
    float cnt = (float)NT;
    float m = s / cnt;
    float var = q / cnt - m*m;
    zstats[col*2]   = m;
    zstats[col*2+1] = rsqrtf(var + 1e-5f);
}

// D3: recompute Z, stage into LDS, apply shift_out permutation + BN + residual + relu.
__global__ __launch_bounds__(128) void kD3(
    const float* __restrict__ x0, const float* __restrict__ Lw, const float* __restrict__ Lb,
    const float* __restrict__ scaleb, const float* __restrict__ zstats,
    const float* __restrict__ g_bn, const float* __restrict__ be_bn,
    float* __restrict__ out)
{
    __shared__ float sLw[64*64];
    __shared__ float sScale[VV*64];
    __shared__ float sX[4][VV*64];
    int tid = threadIdx.x;
    for (int i = tid; i < 64*64; i += 128) sLw[i] = Lw[i];
    for (int i = tid; i < VV*64; i += 128) sScale[i] = scaleb[i];
    __syncthreads();
    int wave = tid >> 5, lane = tid & 31;
    int laneM = lane & 15, half = lane >> 4;
    float* xw = sX[wave];
    int waveg = blockIdx.x * 4 + wave;
    float lb[4];
    #pragma unroll
    for (int jn = 0; jn < 4; ++jn) lb[jn] = Lb[16*jn + laneM];

    for (int it = 0; it < 16; ++it) {
        int nt = waveg + it * D_WAVES;
        int n = nt / TT, t = nt % TT;
        const float* xsrc = x0 + (size_t)n * (CC*TT*VV) + (size_t)t * VV;
        if (it + 1 < 16) {
            int nt2 = waveg + (it+1) * D_WAVES;
            __builtin_prefetch(x0 + (size_t)(nt2/TT)*(CC*TT*VV) + (size_t)(nt2%TT)*VV, 0, 1);
        }
        for (int i2 = lane; i2 < CC*VV; i2 += 32) {
            int cch = i2 / VV, v = i2 % VV;
            xw[i2] = xsrc[(size_t)cch * (TT*VV) + v];
        }
        __syncthreads();
        v8f acc[2][4];
        compute_tile(xw, sLw, sScale, laneM, half, lb, acc);
        __syncthreads();
        // stage Z into xw (reuse) as [v][d]
        #pragma unroll
        for (int i = 0; i < 2; ++i)
            #pragma unroll
            for (int jn = 0; jn < 4; ++jn)
                #pragma unroll
                for (int j = 0; j < 8; ++j) {
                    int v = 16*i + j + 8*half;
                    if (v < VV) xw[v*64 + 16*jn + laneM] = acc[i][jn][j];
                }
        __syncthreads();
        float* outp = out + (size_t)n * (CC*TT*VV) + (size_t)t * VV;
        for (int d = lane; d < 64; d += 32) {
            int dm = d % VV;
            #pragma unroll
            for (int v = 0; v < VV; ++v) {
                int vs = v - dm; if (vs < 0) vs += VV;      // (v-d) mod V
                float zz = xw[vs*64 + d];
                int colS = vs*64 + d;                        // stats follow the source column
                int colG = v*64 + d;                         // gamma/beta at shifted position
                float y = (zz - zstats[colS*2]) * zstats[colS*2+1] * g_bn[colG] + be_bn[colG]
                        + xsrc[(size_t)d * (TT*VV) + v];     // residual x0[n,d,t,v]
                outp[(size_t)d * (TT*VV) + v] = fmaxf(y, 0.f);
            }
        }
        __syncthreads();
    }
}

// ---------------------------------------------------------------------------
extern "C" void kernel_launch(void* const* d_in, const int* in_sizes, int n_in,
                              void* d_out, int out_size, void* d_ws, size_t ws_size,
                              hipStream_t stream)
{
    const float* x0   = (const float*)d_in[0];
    // d_in[1]=shift_in, d_in[2]=shift_out: replaced by closed-form index math
    const float* Wv   = (const float*)d_in[3];
    const float* bv   = (const float*)d_in[4];
    const float* g_v  = (const float*)d_in[5];
    const float* be_v = (const float*)d_in[6];
    const float* Wt   = (const float*)d_in[7];
    const float* bt   = (const float*)d_in[8];
    const float* g_t  = (const float*)d_in[9];
    const float* be_t = (const float*)d_in[10];
    const float* Ws   = (const float*)d_in[11];
    const float* bs   = (const float*)d_in[12];
    const float* g_s  = (const float*)d_in[13];
    const float* be_s = (const float*)d_in[14];
    const float* W2t  = (const float*)d_in[15];
    const float* b2t  = (const float*)d_in[16];
    const float* W2v  = (const float*)d_in[17];
    const float* b2v  = (const float*)d_in[18];
    const float* Lw   = (const float*)d_in[19];
    const float* Lb   = (const float*)d_in[20];
    const float* g_bn = (const float*)d_in[21];
    const float* be_bn= (const float*)d_in[22];
    float* out = (float*)d_out;
    float* ws  = (float*)d_ws;

    float* tconv  = ws;                     // 102400
    float* vconv  = tconv + 102400;         // 1228800
    float* partA  = vconv + 1228800;        // 16384
    float* statsA = partA + 16384;          // 4
    float* xs_raw = statsA + 4;             // 1331200
    float* statsB = xs_raw + 1331200;       // 128
    float* mbuf   = statsB + 128;           // 20800
    float* scaleb = mbuf + 20800;           // 1600
    float* partZ  = scaleb + 1600;          // 1200*1600*2 = 3,840,000
    float* zstats = partZ + 3840000;        // 3200   (total ~26.2 MB)

    kA <<<4096, 256, 0, stream>>>(x0, Wt, bt, Wv, bv, tconv, vconv, partA);
    kA2<<<1,    256, 0, stream>>>(partA, statsA);
    kB <<<dim3(21, 64), 256, 0, stream>>>(tconv, vconv, statsA, g_t, be_t, g_v, be_v, Ws, bs, xs_raw);
    kB2<<<64,   256, 0, stream>>>(xs_raw, statsB);
    kC1<<<325,  64,  0, stream>>>(xs_raw, statsB, g_s, be_s, mbuf);
    kC2<<<75,   256, 0, stream>>>(mbuf, W2t, b2t, out + X0_SZ);
    kC3<<<1,    256, 0, stream>>>(mbuf, W2v, b2v, scaleb);
    kD1<<<D_BLOCKS, 128, 0, stream>>>(x0, Lw, Lb, scaleb, partZ);
    kD2<<<7,    256, 0, stream>>>(partZ, zstats);
    kD3<<<D_BLOCKS, 128, 0, stream>>>(x0, Lw, Lb, scaleb, zstats, g_bn, be_bn, out);
}